// GCN_37606733644141
// MI455X (gfx1250) — compile-verified
//
#include <hip/hip_runtime.h>

// Problem constants (fixed by the reference)
#define DF   128        // feature dim
#define N1C  50000      // middle nodes (not derivable from in_sizes -> hardcode)

typedef float v2f __attribute__((ext_vector_type(2)));
typedef float v8f __attribute__((ext_vector_type(8)));

// ---------------- zero workspace ----------------
__global__ __launch_bounds__(256)
void GCN_zero_kernel(float4* __restrict__ p, int n4) {
    int i = blockIdx.x * blockDim.x + threadIdx.x;
    int stride = gridDim.x * blockDim.x;
    float4 z; z.x = 0.f; z.y = 0.f; z.z = 0.f; z.w = 0.f;
    for (; i < n4; i += stride) p[i] = z;
}

// ---------------- layer 1: edge-weighted SpMM scatter ----------------
// one wave32 per edge; each lane moves 4 consecutive floats (32*4 = 128 = DF)
__global__ __launch_bounds__(256)
void GCN_spmm1_kernel(const float* __restrict__ feat,
                      const float* __restrict__ w,
                      const int*   __restrict__ src,
                      const int*   __restrict__ dst,
                      float*       __restrict__ out,
                      int nE) {
    int e    = (blockIdx.x * blockDim.x + threadIdx.x) >> 5;   // wave-uniform
    int lane = threadIdx.x & 31;
    if (e >= nE) return;
    int u = src[e];
    int v = dst[e];
    float we = w[e];
    const float4* f4 = (const float4*)(feat + (size_t)u * DF);
    float4 m = f4[lane];
    float* o = out + (size_t)v * DF + lane * 4;
    unsafeAtomicAdd(o + 0, m.x * we);
    unsafeAtomicAdd(o + 1, m.y * we);
    unsafeAtomicAdd(o + 2, m.z * we);
    unsafeAtomicAdd(o + 3, m.w * we);
}

// ---------------- layer 2: fused relu(h+b1) gather, weighted scatter ----------------
__global__ __launch_bounds__(256)
void GCN_spmm2_kernel(const float* __restrict__ h,
                      const float* __restrict__ b1,
                      const float* __restrict__ w,
                      const int*   __restrict__ src,
                      const int*   __restrict__ dst,
                      float*       __restrict__ out,
                      int nE) {
    int e    = (blockIdx.x * blockDim.x + threadIdx.x) >> 5;
    int lane = threadIdx.x & 31;
    if (e >= nE) return;
    int u = src[e];
    int v = dst[e];
    float we = w[e];
    const float4* h4 = (const float4*)(h + (size_t)u * DF);
    const float4* bb = (const float4*)(b1);
    float4 m = h4[lane];
    float4 b = bb[lane];
    float r0 = fmaxf(m.x + b.x, 0.f) * we;
    float r1 = fmaxf(m.y + b.y, 0.f) * we;
    float r2 = fmaxf(m.z + b.z, 0.f) * we;
    float r3 = fmaxf(m.w + b.w, 0.f) * we;
    float* o = out + (size_t)v * DF + lane * 4;
    unsafeAtomicAdd(o + 0, r0);
    unsafeAtomicAdd(o + 1, r1);
    unsafeAtomicAdd(o + 2, r2);
    unsafeAtomicAdd(o + 3, r3);
}

// ---------------- dense GEMM: out[M,128] = A[M,128] @ W[128,128] + bias ----------------
// One wave per 16x16 output tile, fp32 WMMA 16x16x4, K-loop of 32 steps.
// Out-of-range A rows are clamped (not zeroed): a clamped row only pollutes
// output rows >= M, which the guarded store drops. This keeps all loads
// branch-free (no exec-mask save/restore inside the K loop).
__global__ __launch_bounds__(256)
void GCN_gemm_wmma_kernel(const float* __restrict__ A,
                          const float* __restrict__ W,
                          const float* __restrict__ bias,
                          float*       __restrict__ out,
                          int M) {
    const int lane = threadIdx.x & 31;
    const int waveInBlk = threadIdx.x >> 5;
    const int nTilesM = (M + 15) >> 4;
    const int nTiles  = nTilesM * (DF / 16);         // N tiles = 8
    int tile = blockIdx.x * (blockDim.x >> 5) + waveInBlk;  // wave-uniform
    if (tile >= nTiles) return;
    const int tM = tile >> 3;
    const int tN = tile & 7;

    const int half = lane >> 4;        // 0: K pair {0,1}; 1: K pair {2,3}
    const int mn   = lane & 15;
    const int row  = tM * 16 + mn;     // A-matrix row for this lane
    const int col  = tN * 16 + mn;     // B/out column for this lane

    // Clamp (see note above) -> loop-invariant base pointers, immediate offsets
    const int rowc = (row < M) ? row : (M - 1);
    const float* Arow = A + (size_t)rowc * DF + 2 * half;        // +k*4 bytes
    const float* Bcol = W + (size_t)(2 * half) * DF + col;       // +k*512 bytes

    v8f c = {};
    #pragma unroll
    for (int k = 0; k < DF; k += 4) {
        const float2 av = *(const float2*)(Arow + k);
        v2f a; a[0] = av.x; a[1] = av.y;
        v2f b;
        b[0] = Bcol[(size_t)k * DF];
        b[1] = Bcol[(size_t)(k + 1) * DF];
        // D = A(16x4) x B(4x16) + C ; emits v_wmma_f32_16x16x4_f32
        c = __builtin_amdgcn_wmma_f32_16x16x4_f32(
                /*neg_a=*/false, a, /*neg_b=*/false, b,
                /*c_mod=*/(short)0, c, /*reuse_a=*/false, /*reuse_b=*/false);
    }

    const float bv = bias[col];
    #pragma unroll
    for (int v = 0; v < 8; ++v) {
        const int r = tM * 16 + v + 8 * half;   // C/D layout: VGPR v -> row v+8*half
        if (r < M) out[(size_t)r * DF + col] = c[v] + bv;
    }
}

extern "C" void kernel_launch(void* const* d_in, const int* in_sizes, int n_in,
                              void* d_out, int out_size, void* d_ws, size_t ws_size,
                              hipStream_t stream) {
    const float* feature = (const float*)d_in[0];
    const float* w0      = (const float*)d_in[1];
    const float* w1      = (const float*)d_in[2];
    const float* b1      = (const float*)d_in[3];
    const float* W2      = (const float*)d_in[4];
    const float* b2      = (const float*)d_in[5];
    const int*   src0    = (const int*)d_in[6];
    const int*   dst0    = (const int*)d_in[7];
    const int*   src1    = (const int*)d_in[8];
    const int*   dst1    = (const int*)d_in[9];

    const int E0 = in_sizes[1];
    const int E1 = in_sizes[2];
    const int N1 = N1C;
    const int N2 = out_size / DF;       // 25000

    float* h   = (float*)d_ws;                 // [N1, DF]
    float* agg = h + (size_t)N1 * DF;          // [N2, DF]
    float* out = (float*)d_out;

    // 1) zero h and agg
    {
        int n4 = (N1 * DF + N2 * DF) / 4;
        int blocks = (n4 + 255) / 256;
        if (blocks > 4096) blocks = 4096;
        GCN_zero_kernel<<<blocks, 256, 0, stream>>>((float4*)d_ws, n4);
    }
    // 2) layer 1 scatter: one wave per edge
    {
        long long threads = (long long)E0 * 32;
        int blocks = (int)((threads + 255) / 256);
        GCN_spmm1_kernel<<<blocks, 256, 0, stream>>>(feature, w0, src0, dst0, h, E0);
    }
    // 3) layer 2 scatter with fused relu(h + b1)
    {
        long long threads = (long long)E1 * 32;
        int blocks = (int)((threads + 255) / 256);
        GCN_spmm2_kernel<<<blocks, 256, 0, stream>>>(h, b1, w1, src1, dst1, agg, E1);
    }
    // 4) WMMA GEMM + bias
    {
        int nTiles = ((N2 + 15) / 16) * (DF / 16);
        int wavesPerBlk = 256 / 32;
        int blocks = (nTiles + wavesPerBlk - 1) / wavesPerBlk;
        GCN_gemm_wmma_kernel<<<blocks, 256, 0, stream>>>(agg, W2, b2, out, N2);
    }
}